// MultiHead_gat_60971355734161
// MI455X (gfx1250) — compile-verified
//
#include <hip/hip_runtime.h>
#include <hip/hip_bf16.h>
#include <math.h>

// ---------------------------------------------------------------------------
// GAT layer for MI455X (gfx1250, wave32).
// GEMMs use V_WMMA_F32_16X16X4_F32 (exact fp32 matrix op, ISA 15.10 op 93).
// Edge softmax-aggregation uses native f32 global atomics (L2-resident: the
// 77MB wv table fits in the 192MB L2, so gathers/atomics never touch HBM).
// ---------------------------------------------------------------------------

typedef __attribute__((ext_vector_type(2))) float v2f;
typedef __attribute__((ext_vector_type(8))) float v8f;

#define F 128
#define H 3
#define HF (H * F)           // 384
#define NEG_SLOPE 0.2f
#define LN_EPS 1e-5f

// ---------------------------------------------------------------------------
// Workspace init (must be re-done every call: harness does not re-poison).
// ---------------------------------------------------------------------------
__global__ void gat_init(float* __restrict__ num, float* __restrict__ denom,
                         float* __restrict__ m, long nnum, long nnh) {
    long i = blockIdx.x * (long)blockDim.x + threadIdx.x;
    long stride = gridDim.x * (long)blockDim.x;
    for (long j = i; j < nnum; j += stride) num[j] = 0.f;
    for (long j = i; j < nnh; j += stride) {
        denom[j] = 0.f;
        m[j] = -__builtin_inff();
    }
}

// ---------------------------------------------------------------------------
// GEMM 1: wv[N,384] = h[N,128] @ W_node^T + b_node     (B[k][j] = W_node[j][k])
// One wave per 16x16 output tile, K=128 -> 32 WMMA f32 16x16x4 per tile.
// A-frag layout (ISA 7.12.2, 32-bit A 16x4): lanes 0-15: M=lane, K={0,1};
// lanes 16-31: M=lane-16, K={2,3}.  D: VGPR r -> M=r (lanes<16) / M=r+8.
// ---------------------------------------------------------------------------
__global__ __launch_bounds__(256)
void gat_gemm_wv(const float* __restrict__ h, const float* __restrict__ W,
                 const float* __restrict__ bn, float* __restrict__ wv, int N) {
    const int lane = threadIdx.x & 31;
    const int wave = threadIdx.x >> 5;
    const int row0 = blockIdx.x * 16;
    const int col0 = blockIdx.y * 128 + wave * 16;   // gridDim.y = 3 -> 384 cols
    const int half = lane >> 4;                      // 0: K pair {0,1}, 1: {2,3}
    const int l15  = lane & 15;                      // A row / B col within tile
    const int koff = half * 2;

    int arow = row0 + l15; if (arow > N - 1) arow = N - 1;
    const float* ap = h + (size_t)arow * F + koff;             // streams a row
    const float* bp = W + (size_t)(col0 + l15) * F + koff;     // streams a row

    v8f acc = {0.f, 0.f, 0.f, 0.f, 0.f, 0.f, 0.f, 0.f};
#pragma unroll 8
    for (int k0 = 0; k0 < F; k0 += 4) {
        v2f a, b;
        a[0] = ap[k0]; a[1] = ap[k0 + 1];
        b[0] = bp[k0]; b[1] = bp[k0 + 1];
        acc = __builtin_amdgcn_wmma_f32_16x16x4_f32(
            false, a, false, b, (short)0, acc, false, false);
    }
    const int col = col0 + l15;
    const float bias = bn[col];
#pragma unroll
    for (int r = 0; r < 8; ++r) {
        int row = row0 + r + half * 8;
        if (row < N) wv[(size_t)row * HF + col] = acc[r] + bias;
    }
}

// ---------------------------------------------------------------------------
// el[n,h] = <wv[n,h,:], att[h,:F]>,  er[n,h] = <wv[n,h,:], att[h,F:]>
// One wave per (node, head); wave32 butterfly reduction.
// ---------------------------------------------------------------------------
__global__ __launch_bounds__(256)
void gat_eler(const float* __restrict__ wv, const float* __restrict__ att,
              float* __restrict__ el, float* __restrict__ er, int N) {
    const int gw = blockIdx.x * 8 + (threadIdx.x >> 5);
    const int lane = threadIdx.x & 31;
    if (gw >= N * H) return;
    const int n = gw / H, hh = gw % H;
    const float* row = wv + (size_t)n * HF + hh * F;
    const float* al = att + hh * 2 * F;
    const float* ar = al + F;
    float sl = 0.f, sr = 0.f;
#pragma unroll
    for (int f = lane; f < F; f += 32) {
        float v = row[f];
        sl += v * al[f];
        sr += v * ar[f];
    }
#pragma unroll
    for (int off = 16; off; off >>= 1) {
        sl += __shfl_xor(sl, off);
        sr += __shfl_xor(sr, off);
    }
    if (lane == 0) { el[gw] = sl; er[gw] = sr; }
}

// ---------------------------------------------------------------------------
// Segment max over edges (per dst, per head).  Float atomic max via the
// sign-split int/uint trick -> native GLOBAL_ATOMIC_MAX_I32 / MIN_U32.
// ---------------------------------------------------------------------------
__global__ void gat_edge_max(const int* __restrict__ src, const int* __restrict__ dst,
                             const float* __restrict__ el, const float* __restrict__ er,
                             float* __restrict__ m, int E) {
    int e = blockIdx.x * blockDim.x + threadIdx.x;
    if (e >= E) return;
    const int s = src[e], d = dst[e];
#pragma unroll
    for (int hh = 0; hh < H; ++hh) {
        float a = el[s * H + hh] + er[d * H + hh];
        a = (a >= 0.f) ? a : NEG_SLOPE * a;
        float* addr = m + d * H + hh;
        if (a >= 0.f) atomicMax((int*)addr, __float_as_int(a));
        else          atomicMin((unsigned int*)addr, __float_as_uint(a));
    }
}

// ---------------------------------------------------------------------------
// Edge scatter: e = exp(attn - m[dst]); denom[dst] += e;
// num[dst,h,:] += e * wv[src,h,:].  Thread = (head, feat); 16 edges/block.
// wv gathers hit L2 (table fits); GLOBAL_ATOMIC_ADD_F32 is native.
// ---------------------------------------------------------------------------
__global__ __launch_bounds__(384)
void gat_edge_scatter(const int* __restrict__ src, const int* __restrict__ dst,
                      const float* __restrict__ wv, const float* __restrict__ el,
                      const float* __restrict__ er, const float* __restrict__ m,
                      float* __restrict__ denom, float* __restrict__ num, int E) {
    const int hh = threadIdx.x >> 7;     // head 0..2
    const int f  = threadIdx.x & 127;    // feature 0..127
    const int e0 = blockIdx.x * 16;
#pragma unroll 1
    for (int i = 0; i < 16; ++i) {
        const int e = e0 + i;
        if (e >= E) break;
        const int s = src[e], d = dst[e];
        float a = el[s * H + hh] + er[d * H + hh];   // broadcast loads
        a = (a >= 0.f) ? a : NEG_SLOPE * a;
        const float w = __expf(a - m[d * H + hh]);
        if (f == 0) atomicAdd(denom + d * H + hh, w);
        atomicAdd(num + (size_t)d * HF + hh * F + f,
                  w * wv[(size_t)s * HF + hh * F + f]);
    }
}

// ---------------------------------------------------------------------------
// agg = num / max(denom, 1e-9)   (in place)
// ---------------------------------------------------------------------------
__global__ void gat_normalize(float* __restrict__ num, const float* __restrict__ denom,
                              long total) {
    long i = blockIdx.x * (long)blockDim.x + threadIdx.x;
    long stride = gridDim.x * (long)blockDim.x;
    for (; i < total; i += stride) {
        long n = i / HF;
        int hh = (int)(i % HF) >> 7;
        num[i] = num[i] / fmaxf(denom[n * H + hh], 1e-9f);
    }
}

// ---------------------------------------------------------------------------
// GEMM 2: hnew[N,128] = agg[N,384] @ w_scale[384,128] + bias
// K=384 -> 96 WMMA f32 16x16x4 per tile; 8 waves cover all 8 col tiles.
// ---------------------------------------------------------------------------
__global__ __launch_bounds__(256)
void gat_gemm_out(const float* __restrict__ agg, const float* __restrict__ Ws,
                  const float* __restrict__ bias, float* __restrict__ hnew, int N) {
    const int lane = threadIdx.x & 31;
    const int wave = threadIdx.x >> 5;
    const int row0 = blockIdx.x * 16;
    const int col0 = wave * 16;                      // 8 waves * 16 = 128 cols
    const int half = lane >> 4;
    const int l15  = lane & 15;
    const int koff = half * 2;

    int arow = row0 + l15; if (arow > N - 1) arow = N - 1;
    const float* ap = agg + (size_t)arow * HF + koff;
    const int col = col0 + l15;

    v8f acc = {0.f, 0.f, 0.f, 0.f, 0.f, 0.f, 0.f, 0.f};
#pragma unroll 4
    for (int k0 = 0; k0 < HF; k0 += 4) {
        v2f a, b;
        a[0] = ap[k0]; a[1] = ap[k0 + 1];
        b[0] = Ws[(size_t)(k0 + koff) * F + col];        // B[k][j] = w_scale[k*F+j]
        b[1] = Ws[(size_t)(k0 + koff + 1) * F + col];
        acc = __builtin_amdgcn_wmma_f32_16x16x4_f32(
            false, a, false, b, (short)0, acc, false, false);
    }
    const float bs = bias[col];
#pragma unroll
    for (int r = 0; r < 8; ++r) {
        int row = row0 + r + half * 8;
        if (row < N) hnew[(size_t)row * F + col] = acc[r] + bs;
    }
}

// ---------------------------------------------------------------------------
// out = relu(LayerNorm(hnew + h) * gamma + beta); one 128-thread block / node.
// ---------------------------------------------------------------------------
__global__ __launch_bounds__(128)
void gat_ln_relu(const float* __restrict__ hnew, const float* __restrict__ h,
                 const float* __restrict__ gamma, const float* __restrict__ beta,
                 float* __restrict__ out, int N) {
    __shared__ float s1[128];
    __shared__ float s2[128];
    const int n = blockIdx.x, f = threadIdx.x;
    const float x = hnew[(size_t)n * F + f] + h[(size_t)n * F + f];
    s1[f] = x;
    __syncthreads();
    for (int st = 64; st > 0; st >>= 1) {
        if (f < st) s1[f] += s1[f + st];
        __syncthreads();
    }
    const float mu = s1[0] * (1.0f / F);
    const float dx = x - mu;
    s2[f] = dx * dx;
    __syncthreads();
    for (int st = 64; st > 0; st >>= 1) {
        if (f < st) s2[f] += s2[f + st];
        __syncthreads();
    }
    const float var = s2[0] * (1.0f / F);
    const float y = dx * rsqrtf(var + LN_EPS) * gamma[f] + beta[f];
    out[(size_t)n * F + f] = fmaxf(y, 0.f);
}

// ---------------------------------------------------------------------------
extern "C" void kernel_launch(void* const* d_in, const int* in_sizes, int n_in,
                              void* d_out, int out_size, void* d_ws, size_t ws_size,
                              hipStream_t stream) {
    const float* h       = (const float*)d_in[0];
    const int*   src     = (const int*)d_in[1];
    const int*   dst     = (const int*)d_in[2];
    const float* W_node  = (const float*)d_in[3];
    const float* b_node  = (const float*)d_in[4];
    const float* att     = (const float*)d_in[5];
    const float* w_scale = (const float*)d_in[6];
    const float* bias    = (const float*)d_in[7];
    const float* gamma   = (const float*)d_in[8];
    const float* beta    = (const float*)d_in[9];

    const int N = in_sizes[0] / F;    // 50000
    const int E = in_sizes[1];        // 800000

    // Workspace layout (floats): wv | num/agg | hnew | el | er | m | denom
    float* ws    = (float*)d_ws;
    float* wv    = ws;                         // N*HF
    float* num   = wv    + (size_t)N * HF;     // N*HF
    float* hnew  = num   + (size_t)N * HF;     // N*F
    float* el    = hnew  + (size_t)N * F;      // N*H
    float* er    = el    + (size_t)N * H;      // N*H
    float* m     = er    + (size_t)N * H;      // N*H
    float* denom = m     + (size_t)N * H;      // N*H

    gat_init<<<1024, 256, 0, stream>>>(num, denom, m, (long)N * HF, (long)N * H);

    dim3 g1((N + 15) / 16, H);
    gat_gemm_wv<<<g1, 256, 0, stream>>>(h, W_node, b_node, wv, N);

    gat_eler<<<(N * H + 7) / 8, 256, 0, stream>>>(wv, att, el, er, N);

    gat_edge_max<<<(E + 255) / 256, 256, 0, stream>>>(src, dst, el, er, m, E);

    gat_edge_scatter<<<(E + 15) / 16, 384, 0, stream>>>(src, dst, wv, el, er, m,
                                                        denom, num, E);

    gat_normalize<<<2048, 256, 0, stream>>>(num, denom, (long)N * HF);

    gat_gemm_out<<<(N + 15) / 16, 256, 0, stream>>>(num, w_scale, bias, hnew, N);

    gat_ln_relu<<<N, 128, 0, stream>>>(hnew, h, gamma, beta, (float*)d_out, N);
}